// JetFFNMoEBlock_11166914970447
// MI455X (gfx1250) — compile-verified
//
#include <hip/hip_runtime.h>
#include <hip/hip_bf16.h>
#include <math.h>
#include <stdint.h>

typedef __attribute__((ext_vector_type(2))) float v2f;
typedef __attribute__((ext_vector_type(4))) float v4f;
typedef __attribute__((ext_vector_type(8))) float v8f;
typedef __attribute__((ext_vector_type(4))) int v4i;
typedef __attribute__((ext_vector_type(4))) unsigned int u32x4;
typedef __attribute__((ext_vector_type(8))) unsigned int u32x8;

#define WMMA_F32(a, b, c) \
    __builtin_amdgcn_wmma_f32_16x16x4_f32(false, (a), false, (b), (short)0, (c), false, false)

#if defined(__gfx1250__) && __has_builtin(__builtin_amdgcn_global_load_async_to_lds_b128)
#define HAVE_ASYNC_LDS 1
#else
#define HAVE_ASYNC_LDS 0
#endif

// ---------------------------------------------------------------------------
// TDM: issue a 2D tile load (rows x cols fp32, global row stride ld elements)
// into LDS at byte offset lds_byte. Builds the D# groups 0/1 per ISA 8.3/8.4
// and issues tensor_load_to_lds via inline asm (portable across toolchains).
// Wave-level op; caller fences with s_wait_tensorcnt + barrier.
// ---------------------------------------------------------------------------
__device__ __forceinline__ void tdm_load_tile_2d(unsigned lds_byte, const float* gsrc,
                                                 unsigned rows, unsigned cols, unsigned ld) {
#if defined(__gfx1250__)
    const unsigned long long ga = (unsigned long long)(uintptr_t)gsrc;
    u32x4 g0;
    g0.x = 1u;                                    // count=1, is_restore=0, gather off
    g0.y = lds_byte;                              // lds_addr
    g0.z = (unsigned)(ga & 0xFFFFFFFFull);        // global_addr[31:0]
    g0.w = (unsigned)((ga >> 32) & 0x01FFFFFFull) // global_addr[56:32]
           | (2u << 30);                          // type=2 (image)
    u32x8 g1;
    g1.s0 = 0x00020000u;                          // wg_mask=0, data_size=2 (4B)
    g1.s1 = (cols & 0xFFFFu) << 16;               // tensor_dim0[15:0]
    g1.s2 = (cols >> 16) | ((rows & 0xFFFFu) << 16); // td0[31:16] | td1[15:0]
    g1.s3 = (rows >> 16) | (cols << 16);          // td1[31:16] | tile_dim0
    g1.s4 = rows & 0xFFFFu;                       // tile_dim1 (tile_dim2=0)
    g1.s5 = ld;                                   // tensor_dim0_stride[31:0]
    g1.s6 = 0u;                                   // stride0[47:32] | stride1[15:0]
    g1.s7 = 0u;                                   // stride1[47:16]
    asm volatile("tensor_load_to_lds %0, %1" :: "s"(g0), "s"(g1) : "memory");
#else
    (void)lds_byte; (void)gsrc; (void)rows; (void)cols; (void)ld;
#endif
}

__device__ __forceinline__ void wait_tensorcnt0() {
#if defined(__gfx1250__)
    __builtin_amdgcn_s_wait_tensorcnt(0);
#endif
}

__device__ __forceinline__ void wait_asynccnt0() {
#if defined(__gfx1250__)
#if __has_builtin(__builtin_amdgcn_s_wait_asynccnt)
    __builtin_amdgcn_s_wait_asynccnt(0);
#else
    asm volatile("s_wait_asynccnt 0x0" ::: "memory");
#endif
#endif
}

// ---------------------------------------------------------------------------
// RMSNorm: one block per token row of H elements
// ---------------------------------------------------------------------------
__global__ __launch_bounds__(256)
void rmsnorm_kernel(const float* __restrict__ x, const float* __restrict__ w,
                    float* __restrict__ out, int H) {
    __shared__ float red[8];
    const int t = blockIdx.x;
    const float* xr = x + (size_t)t * H;
    float ss = 0.f;
    for (int i = threadIdx.x; i < H; i += 256) { float v = xr[i]; ss += v * v; }
    for (int off = 16; off > 0; off >>= 1) ss += __shfl_xor(ss, off, 32);
    if ((threadIdx.x & 31) == 0) red[threadIdx.x >> 5] = ss;
    __syncthreads();
    if (threadIdx.x < 8) {
        float v = red[threadIdx.x];
        v += __shfl_xor(v, 4, 8);
        v += __shfl_xor(v, 2, 8);
        v += __shfl_xor(v, 1, 8);
        if (threadIdx.x == 0) red[0] = v;
    }
    __syncthreads();
    const float inv = rsqrtf(red[0] / (float)H + 1e-6f);
    for (int i = threadIdx.x; i < H; i += 256)
        out[(size_t)t * H + i] = xr[i] * inv * w[i];
}

// ---------------------------------------------------------------------------
// LDS-staged f32 WMMA GEMM. Block = 256 threads (8 waves) computing a
// 64(M) x 128(N) macro tile as a 4x2 grid of 16x64 wave tiles.
// Per 32-wide K chunk:
//   A chunk (64x32, LDS stride 36 for 16B row alignment + conflict-free
//     ds_load_b64) staged via GLOBAL_LOAD_ASYNC_TO_LDS_B128  (ASYNCcnt path)
//   B chunk (32x128, row-major) staged by wave 0 via TDM
//     tensor_load_to_lds                                      (TENSORcnt path)
// Compute: v_wmma_f32_16x16x4_f32 fed entirely from LDS.
// ---------------------------------------------------------------------------
enum { EPI_STORE = 0, EPI_SILU = 1, EPI_ADDSRC = 2, EPI_SCALEACC = 3 };

#define GT_M 64
#define GT_N 128
#define GT_K 32
#define AS_STRIDE 36  // 32 + 4 pad: 144B rows -> 16B aligned, distinct banks

template <int EPI>
__global__ __launch_bounds__(256)
void gemm_wmma_f32(const float* __restrict__ A, const float* __restrict__ B,
                   float* __restrict__ C, const float* __restrict__ aux,
                   int M, int N, int K, int lda, int ldb, int ldc,
                   int scale_stride, int scale_off) {
    __shared__ __align__(16) float As[GT_M * AS_STRIDE];  //  9.2 KB
    __shared__ __align__(16) float Bs[GT_K * GT_N];       // 16.0 KB
    (void)M; (void)N;

    const int tid  = threadIdx.x;
    const int wave = tid >> 5;
    const int lane = tid & 31;
    const int wm = wave & 3;        // 4 wave rows
    const int wn = wave >> 2;       // 2 wave cols
    const int half = lane >> 4;
    const int ncol = lane & 15;
    const int m0 = blockIdx.y * GT_M;
    const int n0 = blockIdx.x * GT_N;

    const unsigned bs_byte =
        (unsigned)(uintptr_t)(__attribute__((address_space(3))) void*)&Bs[0];

    v8f acc0 = {}, acc1 = {}, acc2 = {}, acc3 = {};

    for (int k0 = 0; k0 < K; k0 += GT_K) {
        // ---- stage A chunk: 64 rows x 32 cols = 512 x b128, 2 per thread
#pragma unroll
        for (int it = 0; it < 2; ++it) {
            const int i = tid + 256 * it;
            const int r = i >> 3;          // row 0..63
            const int g = i & 7;           // 16B group within row
            const float* gp = A + (size_t)(m0 + r) * lda + k0 + g * 4;
            float* lp = &As[r * AS_STRIDE + g * 4];
#if HAVE_ASYNC_LDS
            __builtin_amdgcn_global_load_async_to_lds_b128(
                (__attribute__((address_space(1))) v4i*)gp,
                (__attribute__((address_space(3))) v4i*)lp, 0, 0);
#else
            *(v4f*)lp = *(const v4f*)gp;
#endif
        }
        // ---- stage B chunk via TDM (wave 0 issues for whole block)
        if (wave == 0) {
            tdm_load_tile_2d(bs_byte, B + (size_t)k0 * ldb + n0,
                             GT_K, GT_N, (unsigned)ldb);
            wait_tensorcnt0();
        }
#if HAVE_ASYNC_LDS
        wait_asynccnt0();
#endif
        __syncthreads();

        // ---- compute 32 WMMAs per wave from LDS
#pragma unroll
        for (int kk = 0; kk < GT_K; kk += 4) {
            v2f a = *(const v2f*)(&As[(wm * 16 + ncol) * AS_STRIDE + kk + 2 * half]);
            const float* bp = &Bs[(kk + 2 * half) * GT_N + wn * 64 + ncol];
            v2f b;
            b.x = bp[0];  b.y = bp[GT_N];        acc0 = WMMA_F32(a, b, acc0);
            b.x = bp[16]; b.y = bp[GT_N + 16];   acc1 = WMMA_F32(a, b, acc1);
            b.x = bp[32]; b.y = bp[GT_N + 32];   acc2 = WMMA_F32(a, b, acc2);
            b.x = bp[48]; b.y = bp[GT_N + 48];   acc3 = WMMA_F32(a, b, acc3);
        }
        __syncthreads();
    }

    v8f accs[4] = {acc0, acc1, acc2, acc3};
#pragma unroll
    for (int f = 0; f < 4; ++f) {
#pragma unroll
        for (int r = 0; r < 8; ++r) {
            const int m = m0 + wm * 16 + r + 8 * half;
            const int n = n0 + wn * 64 + f * 16 + ncol;
            const size_t idx = (size_t)m * ldc + n;
            float v = accs[f][r];
            if (EPI == EPI_STORE) {
                C[idx] = v;
            } else if (EPI == EPI_SILU) {
                C[idx] = v / (1.0f + __expf(-v));
            } else if (EPI == EPI_ADDSRC) {
                C[idx] = v + aux[idx];
            } else {  // EPI_SCALEACC: C += rowscale * v
                const float s = aux[(size_t)m * scale_stride + scale_off];
                C[idx] += s * v;
            }
        }
    }
}

// ---------------------------------------------------------------------------
// RoPE applied in-place to q,k inside qkv buffer (T, 3H). One thread per
// (token, head, d<32) pair; llama half-split rotation.
// ---------------------------------------------------------------------------
__global__ __launch_bounds__(256)
void rope_kernel(float* __restrict__ qkv, int S, int NH, int T) {
    const int half = 32;  // Dh/2
    const int idx = blockIdx.x * blockDim.x + threadIdx.x;
    const int total = T * NH * half;
    if (idx >= total) return;
    const int d = idx % half;
    const int h = (idx / half) % NH;
    const int t = idx / (half * NH);
    const int s = t % S;
    const float inv_freq = powf(10000.0f, -(float)d / (float)half);
    const float ang = (float)s * inv_freq;
    const float c = cosf(ang), sn = sinf(ang);
    const size_t qb = (size_t)t * 2304 + h * 64;
    float q1 = qkv[qb + d], q2 = qkv[qb + d + half];
    qkv[qb + d]        = q1 * c - q2 * sn;
    qkv[qb + d + half] = q2 * c + q1 * sn;
    const size_t kb = qb + 768;
    float k1 = qkv[kb + d], k2 = qkv[kb + d + half];
    qkv[kb + d]        = k1 * c - k2 * sn;
    qkv[kb + d + half] = k2 * c + k1 * sn;
}

// ---------------------------------------------------------------------------
// Flash attention: one wave per (b, head, 16-row q tile). Online softmax,
// S = Q K^T and O += P V both through v_wmma_f32_16x16x4_f32.
// Row stats (m, l) are uniform across each 16-lane half -> kept in registers.
// P is bounced through LDS to convert C-layout -> A-layout.
// ---------------------------------------------------------------------------
__global__ __launch_bounds__(32)
void flash_attn_kernel(const float* __restrict__ qkv, float* __restrict__ attn_out,
                       int S, int NH) {
    __shared__ float p_smem[16 * 17];
    const int lane = threadIdx.x;
    const int half = lane >> 4;
    const int ncol = lane & 15;
    const int qt = blockIdx.x, h = blockIdx.y, b = blockIdx.z;
    const int qbase = qt * 16;
    const size_t tok0 = (size_t)b * S;

    // Q A-fragments for the whole q tile (Dh=64 -> 16 K-steps of 4)
    const float* Qr = qkv + (tok0 + qbase + ncol) * 2304 + h * 64;
    v2f qa[16];
#pragma unroll
    for (int j = 0; j < 16; ++j) qa[j] = *(const v2f*)(Qr + 4 * j + 2 * half);

    v8f o0 = {}, o1 = {}, o2 = {}, o3 = {};
    float m_prev[8], l_prev[8];
#pragma unroll
    for (int r = 0; r < 8; ++r) { m_prev[r] = -3.0e38f; l_prev[r] = 0.0f; }
    const float scale = 0.125f;  // 1/sqrt(64)

    for (int kt = 0; kt <= qt; ++kt) {
        const int kbase = kt * 16;
        // ---- S = Q K^T (B fragment: B[d][n] = K[kbase+n][d]; contiguous in d)
        const float* Kr = qkv + (tok0 + kbase + ncol) * 2304 + 768 + h * 64;
        v8f sc = {};
#pragma unroll
        for (int j = 0; j < 16; ++j) {
            v2f bf = *(const v2f*)(Kr + 4 * j + 2 * half);
            sc = WMMA_F32(qa[j], bf, sc);
        }
        // ---- online softmax
        const bool diag = (kt == qt);
        float p[8], alpha[8];
#pragma unroll
        for (int r = 0; r < 8; ++r) {
            float v = sc[r] * scale;
            if (diag && (kbase + ncol) > (qbase + r + 8 * half)) v = -3.0e38f;
            float rm = v;
            rm = fmaxf(rm, __shfl_xor(rm, 1, 32));
            rm = fmaxf(rm, __shfl_xor(rm, 2, 32));
            rm = fmaxf(rm, __shfl_xor(rm, 4, 32));
            rm = fmaxf(rm, __shfl_xor(rm, 8, 32));
            const float mnew = fmaxf(m_prev[r], rm);
            const float pv = __expf(v - mnew);
            float rs = pv;
            rs += __shfl_xor(rs, 1, 32);
            rs += __shfl_xor(rs, 2, 32);
            rs += __shfl_xor(rs, 4, 32);
            rs += __shfl_xor(rs, 8, 32);
            alpha[r] = __expf(m_prev[r] - mnew);
            l_prev[r] = l_prev[r] * alpha[r] + rs;
            m_prev[r] = mnew;
            p[r] = pv;
        }
#pragma unroll
        for (int r = 0; r < 8; ++r) {
            o0[r] *= alpha[r]; o1[r] *= alpha[r];
            o2[r] *= alpha[r]; o3[r] *= alpha[r];
        }
        // ---- transpose P (C-layout) into A-layout via LDS
#pragma unroll
        for (int r = 0; r < 8; ++r) p_smem[(r + 8 * half) * 17 + ncol] = p[r];
        __syncthreads();
        v2f pa[4];
#pragma unroll
        for (int j = 0; j < 4; ++j) {
            pa[j].x = p_smem[ncol * 17 + 4 * j + 2 * half];
            pa[j].y = p_smem[ncol * 17 + 4 * j + 2 * half + 1];
        }
        __syncthreads();
        // ---- O += P V
        const float* Vp = qkv + (tok0 + kbase) * 2304 + 1536 + h * 64;
#pragma unroll
        for (int j = 0; j < 4; ++j) {
            const float* vb = Vp + (size_t)(4 * j + 2 * half) * 2304 + ncol;
            v2f bf;
            bf.x = vb[0];  bf.y = vb[2304];  o0 = WMMA_F32(pa[j], bf, o0);
            bf.x = vb[16]; bf.y = vb[2320];  o1 = WMMA_F32(pa[j], bf, o1);
            bf.x = vb[32]; bf.y = vb[2336];  o2 = WMMA_F32(pa[j], bf, o2);
            bf.x = vb[48]; bf.y = vb[2352];  o3 = WMMA_F32(pa[j], bf, o3);
        }
    }
    // ---- finalize
    float* Op = attn_out + (tok0 + qbase) * 768 + h * 64;
#pragma unroll
    for (int r = 0; r < 8; ++r) {
        const float invl = 1.0f / l_prev[r];
        float* orow = Op + (size_t)(r + 8 * half) * 768 + ncol;
        orow[0]  = o0[r] * invl;
        orow[16] = o1[r] * invl;
        orow[32] = o2[r] * invl;
        orow[48] = o3[r] * invl;
    }
}

// ---------------------------------------------------------------------------
// Router: logits -> softmax(8) -> top2 -> normalized combine weights,
// plus atomic accumulators for the switch aux loss.
// ---------------------------------------------------------------------------
__global__ __launch_bounds__(256)
void router_kernel(const float* __restrict__ hn, const float* __restrict__ rw,
                   float* __restrict__ combine, float* __restrict__ auxacc,
                   int T, int H) {
    const int t = blockIdx.x * blockDim.x + threadIdx.x;
    if (t >= T) return;
    float logits[8];
#pragma unroll
    for (int e = 0; e < 8; ++e) logits[e] = 0.f;
    const float* xr = hn + (size_t)t * H;
    for (int i = 0; i < H; ++i) {
        const float xv = xr[i];
        const float* wr = rw + (size_t)i * 8;
#pragma unroll
        for (int e = 0; e < 8; ++e) logits[e] += xv * wr[e];
    }
    float mx = logits[0];
#pragma unroll
    for (int e = 1; e < 8; ++e) mx = fmaxf(mx, logits[e]);
    float probs[8], sum = 0.f;
#pragma unroll
    for (int e = 0; e < 8; ++e) { probs[e] = __expf(logits[e] - mx); sum += probs[e]; }
    const float inv = 1.0f / sum;
#pragma unroll
    for (int e = 0; e < 8; ++e) probs[e] *= inv;
    int i0 = 0;
#pragma unroll
    for (int e = 1; e < 8; ++e) if (probs[e] > probs[i0]) i0 = e;
    int i1 = (i0 == 0) ? 1 : 0;
#pragma unroll
    for (int e = 0; e < 8; ++e) if (e != i0 && probs[e] > probs[i1]) i1 = e;
    const float wsum = probs[i0] + probs[i1];
#pragma unroll
    for (int e = 0; e < 8; ++e) combine[(size_t)t * 8 + e] = 0.f;
    combine[(size_t)t * 8 + i0] = probs[i0] / wsum;
    combine[(size_t)t * 8 + i1] = probs[i1] / wsum;
    atomicAdd(&auxacc[i0], 1.0f);
    atomicAdd(&auxacc[i1], 1.0f);
#pragma unroll
    for (int e = 0; e < 8; ++e) atomicAdd(&auxacc[8 + e], probs[e]);
}

__global__ void aux_finalize_kernel(const float* __restrict__ auxacc,
                                    float* __restrict__ out_aux, int T) {
    if (threadIdx.x == 0 && blockIdx.x == 0) {
        float a = 0.f;
        for (int e = 0; e < 8; ++e) {
            const float frac = auxacc[e] / ((float)T * 2.0f);
            const float mp = auxacc[8 + e] / (float)T;
            a += frac * mp;
        }
        *out_aux = 8.0f * a;
    }
}

__global__ void zero_kernel(float* __restrict__ p, int n) {
    const int i = blockIdx.x * blockDim.x + threadIdx.x;
    if (i < n) p[i] = 0.f;
}

// ---------------------------------------------------------------------------
extern "C" void kernel_launch(void* const* d_in, const int* in_sizes, int n_in,
                              void* d_out, int out_size, void* d_ws, size_t ws_size,
                              hipStream_t stream) {
    (void)in_sizes; (void)n_in; (void)out_size; (void)ws_size;
    const float* x           = (const float*)d_in[0];
    const float* attn_norm_w = (const float*)d_in[1];
    const float* wqkv        = (const float*)d_in[2];
    const float* wo          = (const float*)d_in[3];
    const float* ffn_norm_w  = (const float*)d_in[4];
    const float* router_w    = (const float*)d_in[5];
    const float* w1          = (const float*)d_in[6];
    const float* w2          = (const float*)d_in[7];

    const int B = 2, S = 2048, H = 768, E = 8, F = 3072, NH = 12;
    const int T = B * S;

    float* ws = (float*)d_ws;
    size_t o = 0;
    float* hn      = ws + o; o += (size_t)T * H;       // 12 MB (reused as hn2)
    float* qkv     = ws + o; o += (size_t)T * 3 * H;   // 36 MB
    float* attn    = ws + o; o += (size_t)T * H;       // 12 MB
    float* h1      = ws + o; o += (size_t)T * F;       // 48 MB
    float* combine = ws + o; o += (size_t)T * E;
    float* auxacc  = ws + o; o += 16;

    float* xout    = (float*)d_out;                    // (B,S,H)
    float* aux_out = xout + (size_t)T * H;             // scalar aux

    zero_kernel<<<1, 32, 0, stream>>>(auxacc, 16);

    // x -> rmsnorm -> hn
    rmsnorm_kernel<<<T, 256, 0, stream>>>(x, attn_norm_w, hn, H);
    // qkv = hn @ wqkv   (4096 x 2304 x 768)
    gemm_wmma_f32<EPI_STORE><<<dim3(3 * H / GT_N, T / GT_M), 256, 0, stream>>>(
        hn, wqkv, qkv, nullptr, T, 3 * H, H, H, 3 * H, 3 * H, 0, 0);
    // RoPE on q,k
    rope_kernel<<<(T * NH * 32 + 255) / 256, 256, 0, stream>>>(qkv, S, NH, T);
    // causal flash attention -> attn
    flash_attn_kernel<<<dim3(S / 16, NH, B), 32, 0, stream>>>(qkv, attn, S, NH);
    // x2 = attn @ wo + x   -> d_out
    gemm_wmma_f32<EPI_ADDSRC><<<dim3(H / GT_N, T / GT_M), 256, 0, stream>>>(
        attn, wo, xout, x, T, H, H, H, H, H, 0, 0);
    // hn2 = rmsnorm(x2)  (reuse hn buffer)
    rmsnorm_kernel<<<T, 256, 0, stream>>>(xout, ffn_norm_w, hn, H);
    // routing
    router_kernel<<<(T + 255) / 256, 256, 0, stream>>>(hn, router_w, combine, auxacc, T, H);
    // dense MoE: per expert, h1 = silu(hn2 @ w1[e]); d_out += combine[:,e] * (h1 @ w2[e])
    for (int e = 0; e < E; ++e) {
        const float* w1e = w1 + (size_t)e * H * F;
        const float* w2e = w2 + (size_t)e * F * H;
        gemm_wmma_f32<EPI_SILU><<<dim3(F / GT_N, T / GT_M), 256, 0, stream>>>(
            hn, w1e, h1, nullptr, T, F, H, H, F, F, 0, 0);
        gemm_wmma_f32<EPI_SCALEACC><<<dim3(H / GT_N, T / GT_M), 256, 0, stream>>>(
            h1, w2e, xout, combine, T, H, F, F, H, H, E, e);
    }
    aux_finalize_kernel<<<1, 32, 0, stream>>>(auxacc, aux_out, T);
}